// RotProjNet_15358803050971
// MI455X (gfx1250) — compile-verified
//
#include <hip/hip_runtime.h>

typedef float v2f __attribute__((ext_vector_type(2)));
typedef float v8f __attribute__((ext_vector_type(8)));

#define IM_SIZE 64
#define NROT    24
#define NPB     16384      // points per batch image
#define GROUPS  8          // 16-point groups handled per wave

__global__ void RotProj_zero_kernel(float4* __restrict__ out, int n4) {
    int i = blockIdx.x * blockDim.x + threadIdx.x;
    int stride = gridDim.x * blockDim.x;
    for (; i < n4; i += stride) out[i] = make_float4(0.f, 0.f, 0.f, 0.f);
}

__global__ __launch_bounds__(256) void RotProj_wmma_kernel(
        const float* __restrict__ xyz,   // [32,16384,3]
        const float* __restrict__ rot,   // [24,3,3]
        float* __restrict__ out)         // [32,24,64,64]
{
    const int lane = threadIdx.x & 31;
    // Force wave-uniform values into SGPRs so the image base stays scalar and
    // stores lower to saddr + 32-bit voffset (scale_offset) form.
    const int wave  = __builtin_amdgcn_readfirstlane(
                          (int)((blockIdx.x * blockDim.x + threadIdx.x) >> 5));
    const int base0 = wave * (16 * GROUPS);          // 128 points per wave
    const int b     = base0 / NPB;                   // uniform per wave (16384 % 128 == 0)
    float* __restrict__ imgb = out + (size_t)b * NROT * IM_SIZE * IM_SIZE;

    // --- B matrices (4x16, K x N), loop invariant ---
    // vgpr0 = K0 coeff (lanes 0-15) / K2 coeff (lanes 16-31); vgpr1 = K1/K3 = 0.
    // X tiles produce pixel-x directly:  16*(c_r*x - s_r*z) + 32   (C bias = 32)
    // Z tiles produce the store value:   0.1*(s_r*x + c_r*z)       (C bias = 0)
    const int rA  = lane & 15;        // rotations 0..15  (tile pair 0)
    const int rB  = rA + 16;          // rotations 16..23 (tile pair 1; cols 8..15 padded)
    const int off = (lane < 16) ? 0 : 2;
    v2f BX0 = { 16.0f * rot[rA * 9 + off],     0.0f };
    v2f BZ0 = { 0.1f  * rot[rA * 9 + 6 + off], 0.0f };
    v2f BX1 = { (rB < NROT) ? 16.0f * rot[rB * 9 + off]     : 0.0f, 0.0f };
    v2f BZ1 = { (rB < NROT) ? 0.1f  * rot[rB * 9 + 6 + off] : 0.0f, 0.0f };

    const v8f cX = {32.f, 32.f, 32.f, 32.f, 32.f, 32.f, 32.f, 32.f};
    const v8f c0 = { 0.f,  0.f,  0.f,  0.f,  0.f,  0.f,  0.f,  0.f};

    const int m      = lane & 15;               // A-matrix row (point within group)
    const int m0     = (lane < 16) ? 0 : 8;     // D-matrix row base for this lane
    const int rAbase = rA * (IM_SIZE * IM_SIZE);
    const int rBbase = rB * (IM_SIZE * IM_SIZE);

    for (int g = 0; g < GROUPS; ++g) {
        const int base = base0 + g * 16;

        // --- A matrix (16x4): row m = (x, y, z, 0) ---
        const float* p = xyz + (size_t)3 * (base + m);
        const float px_ = p[0], py_ = p[1], pz_ = p[2];
        v2f A = { (lane < 16) ? px_ : pz_, (lane < 16) ? py_ : 0.0f };

        // Per-point y code, computed once in the lane that owns point m:
        //   okY ? py*64 : -1   (py = rint(16*y + 32), exact-equivalent to ref)
        const int ipy  = (int)rintf(fmaf(py_, 16.0f, 32.0f));
        const int code = ((unsigned)ipy < (unsigned)IM_SIZE) ? (ipy << 6) : -1;

        // D[row=point][col=rotation]; each lane holds matched (pixel-x, value)
        v8f dX0 = __builtin_amdgcn_wmma_f32_16x16x4_f32(false, A, false, BX0, (short)0, cX, false, false);
        v8f dZ0 = __builtin_amdgcn_wmma_f32_16x16x4_f32(false, A, false, BZ0, (short)0, c0, false, false);
        v8f dX1 = __builtin_amdgcn_wmma_f32_16x16x4_f32(false, A, false, BX1, (short)0, cX, false, false);
        v8f dZ1 = __builtin_amdgcn_wmma_f32_16x16x4_f32(false, A, false, BZ1, (short)0, c0, false, false);

        // Broadcast y-codes for this lane's 8 rows (full EXEC here; must be
        // done before the divergent tile-1 region).
        int codev[8];
        #pragma unroll
        for (int v = 0; v < 8; ++v)
            codev[v] = __builtin_amdgcn_ds_bpermute((m0 + v) << 2, code);

        // --- tile pair 0: rotations 0..15, all lanes ---
        #pragma unroll
        for (int v = 0; v < 8; ++v) {
            const int  ipx = (int)rintf(dX0[v]);
            const bool ok  = (codev[v] >= 0) && ((unsigned)ipx < (unsigned)IM_SIZE);
            const int  inner = ok ? (codev[v] + ipx) : 0;   // masked -> pixel (0,0)
            imgb[rAbase + inner] = dZ0[v];
        }

        // --- tile pair 1: rotations 16..23 (single exec-guarded block) ---
        if (rB < NROT) {
            #pragma unroll
            for (int v = 0; v < 8; ++v) {
                const int  ipx = (int)rintf(dX1[v]);
                const bool ok  = (codev[v] >= 0) && ((unsigned)ipx < (unsigned)IM_SIZE);
                const int  inner = ok ? (codev[v] + ipx) : 0;
                imgb[rBbase + inner] = dZ1[v];
            }
        }
    }
}

extern "C" void kernel_launch(void* const* d_in, const int* in_sizes, int n_in,
                              void* d_out, int out_size, void* d_ws, size_t ws_size,
                              hipStream_t stream) {
    const float* xyz = (const float*)d_in[0];   // [32,16384,3]
    const float* rot = (const float*)d_in[1];   // [24,3,3]
    float* out = (float*)d_out;                 // [32,24,64,64] = 3,145,728 f32

    // 1) zero the output image (harness poisons it with 0xAA)
    const int n4 = out_size / 4;                // float4 count
    RotProj_zero_kernel<<<1024, 256, 0, stream>>>((float4*)out, n4);

    // 2) WMMA rotate + project + scatter
    // 524288 points; 128 points/wave -> 4096 waves -> 512 blocks of 256
    const int total_points = in_sizes[0] / 3;
    const int waves  = total_points / (16 * GROUPS);
    const int blocks = (waves * 32) / 256;
    RotProj_wmma_kernel<<<blocks, 256, 0, stream>>>(xyz, rot, out);
}